// GCN_30562987278347
// MI455X (gfx1250) — compile-verified
//
#include <hip/hip_runtime.h>

typedef __attribute__((ext_vector_type(2))) float v2f;
typedef __attribute__((ext_vector_type(8))) float v8f;

#define HID 64
#define NGRAPH 128

// ---------------- utility kernels ----------------

__global__ void k_zero(float* __restrict__ p, int n) {
  int i = blockIdx.x * blockDim.x + threadIdx.x;
  if (i < n) p[i] = 0.0f;
}

__global__ void k_deg(const int* __restrict__ dst, float* __restrict__ deg, int E) {
  int e = blockIdx.x * blockDim.x + threadIdx.x;
  if (e < E) unsafeAtomicAdd(&deg[dst[e]], 1.0f);
}

__global__ void k_rsqrt(float* __restrict__ d, int n) {
  int i = blockIdx.x * blockDim.x + threadIdx.x;
  if (i < n) d[i] = rsqrtf(d[i] + 1.0f);
}

// ---------------- WMMA fp32 GEMM: Y[N,64] = X[N,Fin] @ W[Fin,64] ----------------
// Block: 256 threads = 8 waves. Each wave computes a 16-row strip x 64 cols
// (4 accumulator tiles of 16x16). K stepped by 4 with V_WMMA_F32_16X16X4_F32.
// W is staged into LDS with GLOBAL_LOAD_ASYNC_TO_LDS_B128 (ASYNCcnt-tracked DMA,
// no VGPR round-trip), then B fragments are read via ds_load.
__global__ void __launch_bounds__(256)
k_gemm_wmma(const float* __restrict__ X, const float* __restrict__ W,
            float* __restrict__ Y, int N, int Fin) {
  __shared__ float sW[96 * HID];
  const int tid = threadIdx.x;
  const int total = Fin * HID;            // 6144 (Fin=96) or 4096 (Fin=64): multiple of 1024

  // Async DMA: each thread copies 16B chunks global->LDS. Uniform trip count,
  // EXEC stays all-ones. dsaddr = low 32 bits of the generic LDS pointer.
  for (int i = tid * 4; i < total; i += 1024) {
    unsigned ldsa = (unsigned)(uintptr_t)(&sW[i]);   // LDS byte address
    unsigned goff = (unsigned)(i * 4);               // byte offset from W
    asm volatile("global_load_async_to_lds_b128 %0, %1, %2"
                 :
                 : "v"(ldsa), "v"(goff), "s"(W)
                 : "memory");
  }
  asm volatile("s_wait_asynccnt 0" ::: "memory");
  __syncthreads();

  const int wave = tid >> 5;
  const int lane = tid & 31;
  const int half = lane >> 4;   // 0: lanes 0-15, 1: lanes 16-31
  const int l15  = lane & 15;

  const int row0 = blockIdx.x * 128 + wave * 16;
  int rA = row0 + l15;
  if (rA >= N) rA = N - 1;      // clamp: keeps EXEC all-ones for WMMA
  const float* xr = X + (size_t)rA * Fin + 2 * half;

  v8f acc[4] = {};

  for (int k = 0; k < Fin; k += 4) {
    v2f a = *(const v2f*)(xr + k);        // {X[r][k+2h], X[r][k+2h+1]}
    const int krow = k + 2 * half;
#pragma unroll
    for (int t = 0; t < 4; ++t) {
      v2f b;
      b.x = sW[krow       * HID + t * 16 + l15];
      b.y = sW[(krow + 1) * HID + t * 16 + l15];
      acc[t] = __builtin_amdgcn_wmma_f32_16x16x4_f32(
          false, a, false, b, (short)0, acc[t], false, false);
    }
  }

  // D layout: VGPR v -> row M = v + 8*half, col N = l15 (per 16-col tile)
#pragma unroll
  for (int t = 0; t < 4; ++t) {
#pragma unroll
    for (int v = 0; v < 8; ++v) {
      int r = row0 + half * 8 + v;
      if (r < N) Y[(size_t)r * HID + t * 16 + l15] = acc[t][v];
    }
  }
}

// ---------------- edge scatter: agg[dst] += tmp[src] * dis[src]*dis[dst] ----------------
// 16 lanes per edge, float4 per lane (64 feats). agg fits in L2 -> fast fp32 atomics.
__global__ void k_scatter(const float* __restrict__ tmp,
                          const int* __restrict__ src, const int* __restrict__ dst,
                          const float* __restrict__ dis,
                          float* __restrict__ agg, int E) {
  long long t = (long long)blockIdx.x * blockDim.x + threadIdx.x;
  long long e = t >> 4;
  if (e >= E) return;
  int f4 = ((int)t & 15) * 4;
  int s = src[e], d = dst[e];
  float norm = dis[s] * dis[d];
  const float4 v = *(const float4*)(tmp + (size_t)s * HID + f4);
  float* pd = agg + (size_t)d * HID + f4;
  unsafeAtomicAdd(pd + 0, v.x * norm);
  unsafeAtomicAdd(pd + 1, v.y * norm);
  unsafeAtomicAdd(pd + 2, v.z * norm);
  unsafeAtomicAdd(pd + 3, v.w * norm);
}

// out = relu(agg + tmp * dis^2 + b)   (agg may alias out; 1 thread per element)
__global__ void k_finalize(const float* __restrict__ agg, const float* __restrict__ tmp,
                           const float* __restrict__ dis, const float* __restrict__ b,
                           float* __restrict__ out, int N) {
  int t = blockIdx.x * blockDim.x + threadIdx.x;
  int i = t >> 6, f = t & 63;
  if (i >= N) return;
  float di = dis[i];
  float v = agg[(size_t)i * HID + f] + tmp[(size_t)i * HID + f] * di * di + b[f];
  out[(size_t)i * HID + f] = fmaxf(v, 0.0f);
}

// ---------------- global mean pool (atomics into per-graph sums) ----------------
__global__ void k_pool(const float* __restrict__ h, const int* __restrict__ batch,
                       float* __restrict__ gsum, float* __restrict__ gcnt, int N) {
  int t = blockIdx.x * blockDim.x + threadIdx.x;
  int i = t >> 6, f = t & 63;
  if (i >= N) return;
  int g = batch[i];
  unsafeAtomicAdd(&gsum[(size_t)g * HID + f], h[(size_t)i * HID + f]);
  if (f == 0) unsafeAtomicAdd(&gcnt[g], 1.0f);
}

// ---------------- MLP head: 64 -> leaky -> 32 -> 2, one thread per graph ----------------
__global__ void k_head(const float* __restrict__ gsum, const float* __restrict__ gcnt,
                       const float* __restrict__ Wl, const float* __restrict__ bl,
                       const float* __restrict__ Wp1, const float* __restrict__ bp1,
                       const float* __restrict__ Wp2, const float* __restrict__ bp2,
                       float* __restrict__ out) {
  int g = threadIdx.x;
  if (g >= NGRAPH) return;
  float inv = 1.0f / fmaxf(gcnt[g], 1.0f);
  float pooled[64];
#pragma unroll
  for (int f = 0; f < 64; ++f) pooled[f] = gsum[g * 64 + f] * inv;
  float z1[64];
  for (int j = 0; j < 64; ++j) {
    float a = bl[j];
    for (int f = 0; f < 64; ++f) a += pooled[f] * Wl[f * 64 + j];
    z1[j] = (a > 0.0f) ? a : 0.01f * a;   // LeakyReLU(0.01)
  }
  float z2[32];
  for (int j = 0; j < 32; ++j) {
    float a = bp1[j];
    for (int f = 0; f < 64; ++f) a += z1[f] * Wp1[f * 32 + j];
    z2[j] = a;                             // no activation before last layer
  }
  for (int o = 0; o < 2; ++o) {
    float a = bp2[o];
    for (int f = 0; f < 32; ++f) a += z2[f] * Wp2[f * 2 + o];
    out[g * 2 + o] = a;
  }
}

// ---------------- launcher ----------------

extern "C" void kernel_launch(void* const* d_in, const int* in_sizes, int n_in,
                              void* d_out, int out_size, void* d_ws, size_t ws_size,
                              hipStream_t stream) {
  const float* x    = (const float*)d_in[0];
  const int*   ei   = (const int*)d_in[1];
  const int*   batch= (const int*)d_in[2];
  const float* W1   = (const float*)d_in[3];
  const float* b1   = (const float*)d_in[4];
  const float* W2   = (const float*)d_in[5];
  const float* b2   = (const float*)d_in[6];
  const float* Wl   = (const float*)d_in[7];
  const float* bl   = (const float*)d_in[8];
  const float* Wp1  = (const float*)d_in[9];
  const float* bp1  = (const float*)d_in[10];
  const float* Wp2  = (const float*)d_in[11];
  const float* bp2  = (const float*)d_in[12];
  float* out = (float*)d_out;

  const int FIN = 96;
  const int N = in_sizes[0] / FIN;      // 100000
  const int E = in_sizes[1] / 2;        // 3200000
  const int* src = ei;
  const int* dst = ei + E;

  // workspace layout (floats): dis[N] | buf1[N*64] | buf2[N*64] | gsum[128*64] | gcnt[128]
  float* ws   = (float*)d_ws;
  float* dis  = ws;
  float* buf1 = dis + N;
  float* buf2 = buf1 + (size_t)N * HID;
  float* gsum = buf2 + (size_t)N * HID;
  float* gcnt = gsum + NGRAPH * HID;

  const int TB = 256;
  const int nNH   = N * HID;                    // 6.4M, fits int
  const long long nE16 = (long long)E * 16;

  // degrees -> dis = rsqrt(deg+1)
  k_zero <<<(N + TB - 1) / TB, TB, 0, stream>>>(dis, N);
  k_deg  <<<(E + TB - 1) / TB, TB, 0, stream>>>(dst, dis, E);
  k_rsqrt<<<(N + TB - 1) / TB, TB, 0, stream>>>(dis, N);

  // layer 1: tmp=buf1 = x@W1 ; agg=buf2 ; h -> buf2
  k_gemm_wmma<<<(N + 127) / 128, 256, 0, stream>>>(x, W1, buf1, N, FIN);
  k_zero     <<<(nNH + TB - 1) / TB, TB, 0, stream>>>(buf2, nNH);
  k_scatter  <<<(int)((nE16 + TB - 1) / TB), TB, 0, stream>>>(buf1, src, dst, dis, buf2, E);
  k_finalize <<<(nNH + TB - 1) / TB, TB, 0, stream>>>(buf2, buf1, dis, b1, buf2, N);

  // layer 2: tmp=buf1 = h@W2 ; agg=buf2 ; h2 -> buf2
  k_gemm_wmma<<<(N + 127) / 128, 256, 0, stream>>>(buf2, W2, buf1, N, HID);
  k_zero     <<<(nNH + TB - 1) / TB, TB, 0, stream>>>(buf2, nNH);
  k_scatter  <<<(int)((nE16 + TB - 1) / TB), TB, 0, stream>>>(buf1, src, dst, dis, buf2, E);
  k_finalize <<<(nNH + TB - 1) / TB, TB, 0, stream>>>(buf2, buf1, dis, b2, buf2, N);

  // mean pool + head
  k_zero<<<(NGRAPH * HID + NGRAPH + TB - 1) / TB, TB, 0, stream>>>(gsum, NGRAPH * HID + NGRAPH);
  k_pool<<<(nNH + TB - 1) / TB, TB, 0, stream>>>(buf2, batch, gsum, gcnt, N);
  k_head<<<1, NGRAPH, 0, stream>>>(gsum, gcnt, Wl, bl, Wp1, bp1, Wp2, bp2, out);
}